// AdaptiveFilteringModel_32401233281543
// MI455X (gfx1250) — compile-verified
//
#include <hip/hip_runtime.h>
#include <hip/hip_bf16.h>

typedef __attribute__((ext_vector_type(2))) float v2f;
typedef __attribute__((ext_vector_type(8))) float v8f;

#define XSTR 288              // padded row stride (need 276)
#define NSAMP 262144.0f       // 1024 * 256
#define LRATE 1e-7f

// ---------------------------------------------------------------------------
// Kernel 1: Gram matrix XX = F^T F over 64 features (63 shifted-x features +
// y as feature 63), plus column sums (Xsum, Ysum), via V_WMMA_F32_16X16X4_F32.
// Each block handles 8 batch rows; each of 8 waves handles a 32-sample strip.
// ---------------------------------------------------------------------------
__global__ __launch_bounds__(256) void gram_kernel(const float* __restrict__ in,
                                                   float* __restrict__ gXX,
                                                   float* __restrict__ gSum) {
    __shared__ float xrow[3][XSTR];   // zero-padded rows: xrow[hh][10+w] = x[b,hh,w]
    __shared__ float yrow[256];
    __shared__ float shAcc[64 * 64];
    __shared__ float shSum[64];

    const int tid  = threadIdx.x;
    const int lane = tid & 31;
    const int wv   = tid >> 5;        // wave 0..7
    const int half = lane >> 4;       // lane half selects K pair {0,1} vs {2,3}
    const int lm   = lane & 15;

    for (int i = tid; i < 3 * XSTR; i += 256) (&xrow[0][0])[i] = 0.0f;
    for (int i = tid; i < 64 * 64; i += 256) shAcc[i] = 0.0f;
    if (tid < 64) shSum[tid] = 0.0f;
    __syncthreads();

    v8f acc[4][4];
    #pragma unroll
    for (int a = 0; a < 4; ++a)
        #pragma unroll
        for (int b = 0; b < 4; ++b) acc[a][b] = (v8f){};

    const int b0 = blockIdx.x * 8;
    for (int r = 0; r < 8; ++r) {
        const int bi = b0 + r;
        const float* src = in + (size_t)bi * 1024;   // (4,256) row block

        for (int i = tid; i < 3 * 256; i += 256) {
            int hh = i >> 8, w = i & 255;
            xrow[hh][10 + w] = src[(hh + 1) * 256 + w];
        }
        yrow[tid] = src[tid];
        __syncthreads();

        // --- column-sum partials: thread t covers feature (t&63), chunk (t>>6)
        {
            int f = tid & 63, c = tid >> 6;
            float s = 0.0f;
            int hh = f / 21, j = f - hh * 21;
            const float* p = (f < 63) ? (&xrow[0][0] + hh * XSTR + j) : &yrow[0];
            #pragma unroll 8
            for (int w = 0; w < 64; ++w) s += p[c * 64 + w];
            atomicAdd(&shSum[f], s);
        }

        // --- WMMA Gram accumulation: 8 K-chunks of 4 samples per wave
        for (int kc = 0; kc < 8; ++kc) {
            const int w0 = wv * 32 + kc * 4;
            v2f frag[4];
            #pragma unroll
            for (int t = 0; t < 4; ++t) {
                int fm = t * 16 + lm;                 // feature index (0..63)
                int hh = fm / 21, j = fm - hh * 21;
                const float* base =
                    (fm < 63) ? (&xrow[0][0] + hh * XSTR + j) : &yrow[0];
                // A(16x4) layout: lanes 0-15 hold K=0,1 ; lanes 16-31 hold K=2,3
                frag[t].x = base[w0 + 2 * half];
                frag[t].y = base[w0 + 2 * half + 1];
            }
            #pragma unroll
            for (int tm = 0; tm < 4; ++tm)
                #pragma unroll
                for (int tn = 0; tn < 4; ++tn)
                    acc[tm][tn] = __builtin_amdgcn_wmma_f32_16x16x4_f32(
                        false, frag[tm], false, frag[tn],
                        (short)0, acc[tm][tn], false, false);
        }
        __syncthreads();
    }

    // --- reduce wave accumulators: C(16x16) layout -> LDS -> global atomics
    #pragma unroll
    for (int tm = 0; tm < 4; ++tm)
        #pragma unroll
        for (int tn = 0; tn < 4; ++tn)
            #pragma unroll
            for (int rr = 0; rr < 8; ++rr) {
                int m = tm * 16 + rr + 8 * half;
                int n = tn * 16 + lm;
                atomicAdd(&shAcc[m * 64 + n], acc[tm][tn][rr]);
            }
    __syncthreads();
    for (int i = tid; i < 64 * 64; i += 256) atomicAdd(&gXX[i], shAcc[i]);
    if (tid < 64) atomicAdd(&gSum[tid], shSum[tid]);
}

// ---------------------------------------------------------------------------
// Kernel 2: 500 gradient-descent steps on the reduced statistics.
// loss = mean_b 256*sum_n e^2  =>  dL/dy_pred = -0.5 * e  (Parseval).
// C[f] = sum e * F[:,f] = G[f] - c0*Xsum[f] - (XX ck)[f]
// ---------------------------------------------------------------------------
__global__ __launch_bounds__(64) void train_kernel(const float* __restrict__ gXX,
                                                   const float* __restrict__ gSum,
                                                   const float* __restrict__ k1_in,
                                                   const float* __restrict__ b1_in,
                                                   const float* __restrict__ k2_in,
                                                   const float* __restrict__ b2_in,
                                                   float* __restrict__ gCk,
                                                   float* __restrict__ gC0) {
    __shared__ float XXs[64 * 64];
    __shared__ float xs[64];          // xs[63] = Ysum
    __shared__ float k1s[64];
    __shared__ float k2s[4];
    __shared__ float b1s, b2s;
    __shared__ float cks[64];
    __shared__ float Cs[64];
    __shared__ float ps[64];
    __shared__ float Es;
    __shared__ float gk2s[4], gb1s, gb2s;

    const int t = threadIdx.x;
    for (int i = t; i < 4096; i += 64) XXs[i] = gXX[i];
    xs[t] = gSum[t];
    if (t < 63) k1s[t] = k1_in[t];
    if (t == 63) { k1s[63] = 0.0f; b1s = b1_in[0]; b2s = b2_in[0]; }
    if (t < 3) k2s[t] = k2_in[t];
    __syncthreads();

    for (int it = 0; it < 500; ++it) {
        // combined kernel ck[hh,j] = sum_i k2[i]*k1[hh-i+1, j]
        if (t < 63) {
            int hh = t / 21, j = t - hh * 21;
            float s = 0.0f;
            for (int i = 0; i < 3; ++i) {
                int ip = hh - i + 1;
                if (ip >= 0 && ip <= 2) s += k2s[i] * k1s[ip * 21 + j];
            }
            cks[t] = s;
        } else {
            cks[63] = 0.0f;
        }
        __syncthreads();

        float S2 = k2s[0] + k2s[1] + k2s[2];
        float c0 = b2s + b1s * S2;

        // C[f] = G[f] - c0*Xsum[f] - (XX ck)[f]   (G[f] == XX[f][63])
        {
            float u = 0.0f;
            const float* row = &XXs[t * 64];
            #pragma unroll 8
            for (int f2 = 0; f2 < 64; ++f2) u += row[f2] * cks[f2];
            Cs[t] = XXs[t * 64 + 63] - c0 * xs[t] - u;
            ps[t] = xs[t] * cks[t];
        }
        __syncthreads();

        if (t == 0) {
            float e = xs[63] - NSAMP * c0;          // Ysum - N*c0
            for (int f = 0; f < 64; ++f) e -= ps[f];
            Es = e;
        }
        __syncthreads();

        // gradients (at current params)
        float gk1 = 0.0f;
        if (t < 63) {
            int ip = t / 21, j = t - ip * 21;
            for (int h = 0; h < 3; ++h) {
                int hh = h + ip - 1;
                if (hh >= 0 && hh <= 2) gk1 += k2s[h] * Cs[hh * 21 + j];
            }
            gk1 *= -0.5f;
        }
        if (t == 0) {
            float E = Es;
            for (int h = 0; h < 3; ++h) {
                float s = b1s * E;
                for (int f = 0; f < 63; ++f) {
                    int ip = f / 21, j = f - ip * 21;
                    int hh = h + ip - 1;
                    if (hh >= 0 && hh <= 2) s += k1s[f] * Cs[hh * 21 + j];
                }
                gk2s[h] = -0.5f * s;
            }
            gb1s = -0.5f * Es * S2;
            gb2s = -0.5f * Es;
        }
        __syncthreads();

        if (t < 63) k1s[t] -= LRATE * gk1;
        if (t == 0) {
            for (int h = 0; h < 3; ++h) k2s[h] -= LRATE * gk2s[h];
            b1s -= LRATE * gb1s;
            b2s -= LRATE * gb2s;
        }
        __syncthreads();
    }

    // emit final combined kernel + bias
    if (t < 63) {
        int hh = t / 21, j = t - hh * 21;
        float s = 0.0f;
        for (int i = 0; i < 3; ++i) {
            int ip = hh - i + 1;
            if (ip >= 0 && ip <= 2) s += k2s[i] * k1s[ip * 21 + j];
        }
        gCk[t] = s;
    }
    if (t == 63) gCk[63] = 0.0f;
    if (t == 0) gC0[0] = b2s + b1s * (k2s[0] + k2s[1] + k2s[2]);
}

// ---------------------------------------------------------------------------
// Kernel 3: out[b,w] = y[b,w] - c0 - sum_{hh,j} ck[hh,j]*xp[b,hh,w+j]
// ---------------------------------------------------------------------------
__global__ __launch_bounds__(256) void predict_kernel(const float* __restrict__ in,
                                                      const float* __restrict__ gCk,
                                                      const float* __restrict__ gC0,
                                                      float* __restrict__ out) {
    __shared__ float xrow[3][XSTR];
    __shared__ float cks[64];
    const int tid = threadIdx.x;
    const int bi  = blockIdx.x;

    for (int i = tid; i < 3 * XSTR; i += 256) (&xrow[0][0])[i] = 0.0f;
    if (tid < 64) cks[tid] = gCk[tid];
    __syncthreads();

    const float* src = in + (size_t)bi * 1024;
    for (int i = tid; i < 3 * 256; i += 256) {
        int hh = i >> 8, w = i & 255;
        xrow[hh][10 + w] = src[(hh + 1) * 256 + w];
    }
    __syncthreads();

    float accv = src[tid] - gC0[0];
    #pragma unroll
    for (int hh = 0; hh < 3; ++hh) {
        const float* p = &xrow[hh][tid];
        #pragma unroll
        for (int j = 0; j < 21; ++j) accv -= cks[hh * 21 + j] * p[j];
    }
    out[(size_t)bi * 256 + tid] = accv;
}

// ---------------------------------------------------------------------------
extern "C" void kernel_launch(void* const* d_in, const int* in_sizes, int n_in,
                              void* d_out, int out_size, void* d_ws, size_t ws_size,
                              hipStream_t stream) {
    const float* in  = (const float*)d_in[0];   // (1024,4,256,1)
    const float* k1  = (const float*)d_in[1];   // (3,21,1,1)
    const float* b1  = (const float*)d_in[2];   // (1,)
    const float* k2  = (const float*)d_in[3];   // (3,1,1,1)
    const float* b2  = (const float*)d_in[4];   // (1,)
    float* out = (float*)d_out;                 // (1024,256)
    float* ws  = (float*)d_ws;

    float* gXX  = ws;           // 4096 floats (64x64 Gram, feat 63 = y)
    float* gSum = ws + 4096;    // 64 floats   (Xsum[0..62], Ysum)
    float* gCk  = ws + 4160;    // 64 floats   (final combined kernel)
    float* gC0  = ws + 4224;    // 1 float     (final combined bias)

    hipMemsetAsync(ws, 0, 4160 * sizeof(float), stream);
    gram_kernel<<<128, 256, 0, stream>>>(in, gXX, gSum);
    train_kernel<<<1, 64, 0, stream>>>(gXX, gSum, k1, b1, k2, b2, gCk, gC0);
    predict_kernel<<<1024, 256, 0, stream>>>(in, gCk, gC0, out);
}